// LQE_21698174779947
// MI455X (gfx1250) — compile-verified
//
#include <hip/hip_runtime.h>
#include <stdint.h>

typedef __attribute__((ext_vector_type(2))) float v2f;
typedef __attribute__((ext_vector_type(4))) float v4f;
typedef __attribute__((ext_vector_type(8))) float v8f;
typedef __attribute__((ext_vector_type(4))) int   v4i;

#define NWAVES 4
#define BLOCK  (NWAVES * 32)
#define CORN16 (132 * 16)              // 2112 floats: 16 anchors x 4 sides x 33 logits
#define STAT16 (20 * 16)               // 320 floats : 16 anchors x 20 stat features
#define SLICE  (CORN16 + STAT16 + 16)  // + 16 quality scalars = 2448 floats/wave

// CDNA5 async L2->LDS path (direct, ASYNCcnt-tracked); sync b128 copy fallback.
#if defined(__has_builtin)
#if __has_builtin(__builtin_amdgcn_global_load_async_to_lds_b128) && \
    __has_builtin(__builtin_amdgcn_s_wait_asynccnt)
#define ASYNC_LDS 1
#endif
#endif

// Fused LQE head: softmax -> top4 -> [16x20]x[20x64] WMMA MLP -> SiLU -> dot(w2)
// -> broadcast add onto 80 score channels. One pass over HBM (memory-bound,
// ~935 MB => ~40us floor at 23.3 TB/s; WMMA makes the MLP free).
__global__ __launch_bounds__(BLOCK) void lqe_fused(
    const float* __restrict__ scores,    // [nA, 80]
    const float* __restrict__ corners,   // [nA, 132]
    const float* __restrict__ w1,        // [20, 64] row-major
    const float* __restrict__ b1,        // [64]
    const float* __restrict__ w2,        // [64, 1]
    const float* __restrict__ b2,        // [1]
    float* __restrict__ out,             // [nA, 80]
    int nA, int nGroups)
{
    __shared__ float smem[NWAVES * SLICE];
    const int lane = threadIdx.x & 31;
    const int wid  = threadIdx.x >> 5;
    float* corn = smem + wid * SLICE;
    float* stat = corn + CORN16;
    float* qual = stat + STAT16;

    const int hf = lane >> 4;   // which half of the wave (K-pair select)
    const int nl = lane & 15;   // N / M index within half

    // ---- persistent weight fragments in VGPRs (reused across all groups) ----
    // B (w1, 4x16 per kc/nb tile): lanes 0-15 hold rows K=k0,k0+1; lanes 16-31 rows k0+2,k0+3.
    v2f bfrag[5][4];
#pragma unroll
    for (int kc = 0; kc < 5; ++kc)
#pragma unroll
        for (int nb = 0; nb < 4; ++nb) {
            const int k0 = 4 * kc + 2 * hf;
            const int n  = nb * 16 + nl;
            v2f f;
            f.x = w1[(k0 + 0) * 64 + n];
            f.y = w1[(k0 + 1) * 64 + n];
            bfrag[kc][nb] = f;
        }
    float b1v[4], w2v[4];
#pragma unroll
    for (int nb = 0; nb < 4; ++nb) {
        b1v[nb] = b1[nb * 16 + nl];
        w2v[nb] = w2[nb * 16 + nl];
    }
    const float b2v = b2[0];

    const int gw      = blockIdx.x * NWAVES + wid;
    const int gstride = gridDim.x * NWAVES;

    for (int g = gw; g < nGroups; g += gstride) {
        const long a0 = (long)g * 16;

        // speculative prefetch of the next group's corner tile (8448 B / group)
        {
            const long gn = (long)g + gstride;
            if (gn < nGroups)
                __builtin_prefetch(corners + gn * (16L * 132) + lane * 64, 0, 1);
        }

        // ---- stage 16 anchors x 132 floats into LDS (anchor = exactly 33 float4) ----
#pragma unroll
        for (int it = 0; it < 17; ++it) {
            const int i = it * 32 + lane;
            if (i < 528) {
                const int al = i / 33;               // local anchor 0..15
                long ai = a0 + al;
                if (ai > (long)nA - 1) ai = (long)nA - 1;  // clamp for ragged tail
                const v4f* gp = &((const v4f*)corners)[ai * 33 + (i - al * 33)];
                v4f* lp = &((v4f*)corn)[i];
#ifdef ASYNC_LDS
                // direct L2->LDS b128, ASYNCcnt-tracked (no VGPR round trip)
                typedef __attribute__((address_space(1))) v4i* gas1_t;
                typedef __attribute__((address_space(3))) v4i* las3_t;
                gas1_t ga = (gas1_t)(uintptr_t)gp;
                las3_t la = (las3_t)(uint32_t)(uintptr_t)lp;   // low 32b == LDS offset
                __builtin_amdgcn_global_load_async_to_lds_b128(ga, la, 0, 0);
#else
                *lp = *gp;
#endif
            }
        }
#ifdef ASYNC_LDS
        __builtin_amdgcn_s_wait_asynccnt(0);
#endif
        __builtin_amdgcn_wave_barrier();

        // ---- per-side softmax + top-4 (64 side tasks, 2 passes of 32 lanes) ----
#pragma unroll
        for (int p = 0; p < 2; ++p) {
            const int task = p * 32 + lane;          // anchor = task>>2, side = task&3
            const float* s = corn + task * 33;       // 33-word stride: bank-conflict-free
            float t0 = -3.4e38f, t1 = -3.4e38f, t2 = -3.4e38f, t3 = -3.4e38f;
            for (int j = 0; j < 33; ++j) {           // branchless 4-max insertion
                float v  = s[j];
                float hi = fmaxf(t0, v); v = fminf(t0, v); t0 = hi;
                hi = fmaxf(t1, v); v = fminf(t1, v); t1 = hi;
                hi = fmaxf(t2, v); v = fminf(t2, v); t2 = hi;
                t3 = fmaxf(t3, v);
            }
            float sum = 0.f;
            for (int j = 0; j < 33; ++j) sum += __expf(s[j] - t0);
            const float inv = __builtin_amdgcn_rcpf(sum);   // v_rcp_f32, no div fixup
            const float e0 = inv;                           // exp(t0 - t0) = 1
            const float e1 = __expf(t1 - t0) * inv;
            const float e2 = __expf(t2 - t0) * inv;
            const float e3 = __expf(t3 - t0) * inv;
            float* st = stat + task * 5;
            st[0] = e0; st[1] = e1; st[2] = e2; st[3] = e3;
            st[4] = 0.25f * (e0 + e1 + e2 + e3);
        }
        __builtin_amdgcn_wave_barrier();

        // ---- A fragments (16x20 stat slab) per ISA f32 16x4 layout ----
        v2f afr[5];
#pragma unroll
        for (int kc = 0; kc < 5; ++kc) {
            const int k0 = 4 * kc + 2 * hf;
            v2f f;
            f.x = stat[nl * 20 + k0];
            f.y = stat[nl * 20 + k0 + 1];
            afr[kc] = f;
        }

        // ---- D = stat @ w1 : 4 N-tiles x 5 chained K=4 steps, f32 WMMA ----
        v8f acc[4];
        const v8f zero = {0.f, 0.f, 0.f, 0.f, 0.f, 0.f, 0.f, 0.f};
#pragma unroll
        for (int nb = 0; nb < 4; ++nb) acc[nb] = zero;
#pragma unroll
        for (int kc = 0; kc < 5; ++kc)
#pragma unroll
            for (int nb = 0; nb < 4; ++nb)
                acc[nb] = __builtin_amdgcn_wmma_f32_16x16x4_f32(
                    false, afr[kc], false, bfrag[kc][nb],
                    (short)0, acc[nb], false, false);

        // ---- SiLU + w2 projection; C/D layout: lane n = nl, row m = r + 8*hf ----
        float part[8];
#pragma unroll
        for (int r = 0; r < 8; ++r) part[r] = 0.f;
#pragma unroll
        for (int nb = 0; nb < 4; ++nb)
#pragma unroll
            for (int r = 0; r < 8; ++r) {
                const float hv = acc[nb][r] + b1v[nb];
                const float sg = __builtin_amdgcn_rcpf(1.f + __expf(-hv));
                part[r] += hv * sg * w2v[nb];
            }
        // reduce over the 16 N-lanes of each half-wave
#pragma unroll
        for (int m = 1; m < 16; m <<= 1)
#pragma unroll
            for (int r = 0; r < 8; ++r)
                part[r] += __shfl_xor(part[r], m, 32);
        if (nl == 0) {                               // lanes 0 and 16 publish quality
#pragma unroll
            for (int r = 0; r < 8; ++r)
                qual[hf * 8 + r] = part[r] + b2v;
        }
        __builtin_amdgcn_wave_barrier();

        // ---- broadcast quality onto 80 score channels (20 float4 / anchor) ----
        const v4f* sc4 = (const v4f*)scores;
        v4f* ob4 = (v4f*)out;
#pragma unroll
        for (int it = 0; it < 10; ++it) {
            const int i  = it * 32 + lane;           // 0..319
            const int al = i / 20;                   // local anchor
            const long an = a0 + al;
            if (an < (long)nA) {
                const float q = qual[al];
                v4f v = sc4[an * 20 + (i - al * 20)];
                v.x += q; v.y += q; v.z += q; v.w += q;
                ob4[an * 20 + (i - al * 20)] = v;
            }
        }
    }
}

extern "C" void kernel_launch(void* const* d_in, const int* in_sizes, int n_in,
                              void* d_out, int out_size, void* d_ws, size_t ws_size,
                              hipStream_t stream) {
    const float* scores  = (const float*)d_in[0];
    const float* corners = (const float*)d_in[1];
    const float* w1      = (const float*)d_in[2];
    const float* b1      = (const float*)d_in[3];
    const float* w2      = (const float*)d_in[4];
    const float* b2      = (const float*)d_in[5];
    float* out = (float*)d_out;

    const int nA = in_sizes[1] / 132;                 // 4*(REG_MAX+1) = 132
    const int nGroups = (nA + 15) / 16;
    int blocks = (nGroups + NWAVES - 1) / NWAVES;
    if (blocks > 2048) blocks = 2048;                 // grid-stride persistent waves

    lqe_fused<<<blocks, BLOCK, 0, stream>>>(scores, corners, w1, b1, w2, b2,
                                            out, nA, nGroups);
}